// PreFusionCat_51788715655731
// MI455X (gfx1250) — compile-verified
//
#include <hip/hip_runtime.h>
#include <stdint.h>

#define TH 256
#define TW 256
#define HWSZ (TH * TW)          // 65536 pixels per sample
#define CCH 512                 // channels
#define NBATCH 2
#define BN_EPS 1e-5f

#define BM 128
#define BNT 128
#define BK 16
#define LDPB 136                // B tile row stride (floats): 2 rows apart -> +16 banks
#define LDPA 20                 // A tile row stride (floats): 20*m mod 64 distinct

typedef float v2f __attribute__((ext_vector_type(2)));
typedef float v8f __attribute__((ext_vector_type(8)));
typedef int   vi4 __attribute__((__vector_size__(4 * sizeof(int))));  // matches builtin param

// ---------------------------------------------------------------------------
// gfx1250 async global->LDS path (ASYNCcnt-tracked), with safe fallback.
// Probe-derived signature: (vi4 AS(1)* gsrc, vi4 AS(3)* ldst, imm, imm).
// ---------------------------------------------------------------------------
#if __has_builtin(__builtin_amdgcn_global_load_async_to_lds_b128)
#define ASYNC_LDS 1
#else
#define ASYNC_LDS 0
#endif

typedef __attribute__((address_space(1))) vi4* gptr16_t;
typedef __attribute__((address_space(3))) vi4* lptr16_t;

__device__ __forceinline__ void copy16_g2l(const float* g, float* l) {
#if ASYNC_LDS
  __builtin_amdgcn_global_load_async_to_lds_b128(
      (gptr16_t)(uintptr_t)g,
      (lptr16_t)(uint32_t)(uintptr_t)l,   // low 32 bits of flat addr = LDS offset
      0, 0);
#else
  *reinterpret_cast<float4*>(l) = *reinterpret_cast<const float4*>(g);
#endif
}

__device__ __forceinline__ void wait_async_all() {
#if ASYNC_LDS
#if __has_builtin(__builtin_amdgcn_s_wait_asynccnt)
  __builtin_amdgcn_s_wait_asynccnt(0);
#else
  asm volatile("s_wait_asynccnt 0x0" ::: "memory");
#endif
#endif
}

// ---------------------------------------------------------------------------
// Kernel 1: img_m[n][c][h][w] = sum of 5 FPN levels, nearest-upsampled by
// (n+1) for levels>0, edge-padded to 256x256.  Pure streaming (memory bound).
// ---------------------------------------------------------------------------
__global__ __launch_bounds__(256) void build_img_m(
    const float* __restrict__ f0, const float* __restrict__ f1,
    const float* __restrict__ f2, const float* __restrict__ f3,
    const float* __restrict__ f4, float* __restrict__ out) {
  long idx = (long)blockIdx.x * 256 + threadIdx.x;   // [0, 2*512*65536)
  int w  = (int)(idx & (TW - 1));
  int h  = (int)((idx >> 8) & (TH - 1));
  long cc = idx >> 16;                               // n*C + c
  int n  = (int)(cc >> 9);

  auto samp = [&](const float* __restrict__ f, int size, int s) -> float {
    int up = size * s;                 // upsampled size (both even -> symmetric pad)
    int pt = (TH - up) >> 1;           // dh//2 (== dw//2, square)
    int uh = h - pt; uh = uh < 0 ? 0 : uh; uh = uh > up - 1 ? up - 1 : uh;
    int uw = w - pt; uw = uw < 0 ? 0 : uw; uw = uw > up - 1 ? up - 1 : uw;
    int shf = s - 1;                   // s in {1,2} -> shift 0 or 1
    int sh = uh >> shf, sw = uw >> shf;
    return f[((long)cc * size + sh) * size + sw];
  };

  int s = n + 1;
  float acc = samp(f0, 128, 1)        // level 0 never upsampled
            + samp(f1, 64, s) + samp(f2, 32, s)
            + samp(f3, 16, s) + samp(f4, 8, s);
  out[idx] = acc;
}

// ---------------------------------------------------------------------------
// Kernel 2: fold eval-mode BN + conv bias into per-channel affine.
// ---------------------------------------------------------------------------
__global__ void bn_affine(const float* __restrict__ b_, const float* __restrict__ gamma,
                          const float* __restrict__ beta, const float* __restrict__ mean,
                          const float* __restrict__ var, float* __restrict__ alpha,
                          float* __restrict__ bet) {
  int o = blockIdx.x * blockDim.x + threadIdx.x;
  if (o < CCH) {
    float a = gamma[o] / sqrtf(var[o] + BN_EPS);
    alpha[o] = a;
    bet[o]   = a * (b_[o] - mean[o]) + beta[o];
  }
}

// ---------------------------------------------------------------------------
// Kernel 3: double-buffered async-LDS WMMA f32 GEMM
//   Out[o,p] = relu(alpha[o]*(W[o,:]·X[:,p]) + bet[o])
// M=K=512, N=65536/sample. BM=BN=128, BK=16, 256 threads = 8 wave32s (4Mx2N),
// each wave 2x4 fragments of v_wmma_f32_16x16x4_f32.  Tile for chunk k+1 is
// issued with global_load_async_to_lds_b128 before computing chunk k; the
// s_wait_asynccnt + barrier at chunk end closes both hazards.
// No divergence -> EXEC all ones for every WMMA (ISA requirement).
// ---------------------------------------------------------------------------
__global__ __launch_bounds__(256) void gemm_bn_relu(
    const float* __restrict__ X,      // [C, HW] per sample
    const float* __restrict__ Wm,     // [C, C] row-major (o, c)
    const float* __restrict__ alpha, const float* __restrict__ bet,
    float* __restrict__ Out,          // base incl. branch offset
    long xSampleStride, long outSampleStride) {
  __shared__ float Am[2][BM][LDPA];   // Am[b][m][k] = W[m0+m][k0+k]
  __shared__ float Bs[2][BK][LDPB];   // Bs[b][k][n] = X[k0+k][p0+n]

  const int tid  = threadIdx.x;
  const int lane = tid & 31;
  const int wid  = tid >> 5;
  const int wm   = wid >> 1;          // 0..3  (32 rows each)
  const int wn   = wid & 1;           // 0..1  (64 cols each)

  const long p0 = (long)blockIdx.x * BNT;
  const int  m0 = blockIdx.y * BM;
  const float* Xs = X + (long)blockIdx.z * xSampleStride;
  float*       Os = Out + (long)blockIdx.z * outSampleStride;

  v8f acc[2][4];
#pragma unroll
  for (int mi = 0; mi < 2; ++mi)
#pragma unroll
    for (int ni = 0; ni < 4; ++ni) acc[mi][ni] = (v8f){0.f,0.f,0.f,0.f,0.f,0.f,0.f,0.f};

  const int krow0 = (lane < 16) ? 0 : 2;   // f32 A/B fragment K placement (ISA 7.12.2)
  const int lm    = lane & 15;

  // issue one BK-chunk (A: 128x16, B: 16x128) into LDS buffer `buf`
  auto issue_tiles = [&](int buf, int k0) {
#pragma unroll
    for (int j = 0; j < 2; ++j) {          // B: 512 float4, 2 per thread
      int f  = tid + j * 256;
      int k  = f >> 5;                     // 32 float4 per 128-float row
      int c4 = (f & 31) << 2;
      copy16_g2l(Xs + (long)(k0 + k) * HWSZ + p0 + c4, &Bs[buf][k][c4]);
    }
#pragma unroll
    for (int j = 0; j < 2; ++j) {          // A: 512 float4, 2 per thread
      int f  = tid + j * 256;
      int m  = f >> 2;                     // 4 float4 per 16-float row
      int k4 = (f & 3) << 2;
      copy16_g2l(Wm + (long)(m0 + m) * CCH + k0 + k4, &Am[buf][m][k4]);
    }
  };

  issue_tiles(0, 0);
  wait_async_all();
  __syncthreads();

  for (int k0 = 0; k0 < CCH; k0 += BK) {
    const int cur = (k0 >> 4) & 1;
    if (k0 + BK < CCH) issue_tiles(cur ^ 1, k0 + BK);   // overlap with compute

#pragma unroll
    for (int kk = 0; kk < BK; kk += 4) {
      v2f a[2], b[4];
#pragma unroll
      for (int mi = 0; mi < 2; ++mi) {     // adjacent K pair -> one ds_load_b64
        int m = wm * 32 + mi * 16 + lm;
        a[mi] = *reinterpret_cast<const v2f*>(&Am[cur][m][kk + krow0]);
      }
#pragma unroll
      for (int ni = 0; ni < 4; ++ni) {     // two rows -> ds_load_2addr_b32
        int nn = wn * 64 + ni * 16 + lm;
        b[ni].x = Bs[cur][kk + krow0][nn];
        b[ni].y = Bs[cur][kk + krow0 + 1][nn];
      }
#pragma unroll
      for (int mi = 0; mi < 2; ++mi)
#pragma unroll
        for (int ni = 0; ni < 4; ++ni)
          acc[mi][ni] = __builtin_amdgcn_wmma_f32_16x16x4_f32(
              false, a[mi], false, b[ni], (short)0, acc[mi][ni], false, false);
    }

    wait_async_all();   // next-chunk async copies complete (this wave)
    __syncthreads();    // all waves done reading `cur` + all copies visible
  }

  // ---- epilogue: fused BN affine + ReLU, direct store to output half
  const int mh = (lane < 16) ? 0 : 8;      // D layout: VGPR r -> M = r or r+8
#pragma unroll
  for (int mi = 0; mi < 2; ++mi) {
    int ob = m0 + wm * 32 + mi * 16 + mh;
#pragma unroll
    for (int ni = 0; ni < 4; ++ni) {
      long pp = p0 + wn * 64 + ni * 16 + lm;
#pragma unroll
      for (int r = 0; r < 8; ++r) {
        int o = ob + r;
        float v = acc[mi][ni][r] * alpha[o] + bet[o];
        Os[(long)o * HWSZ + pp] = fmaxf(v, 0.f);
      }
    }
  }
}

// ---------------------------------------------------------------------------
extern "C" void kernel_launch(void* const* d_in, const int* in_sizes, int n_in,
                              void* d_out, int out_size, void* d_ws, size_t ws_size,
                              hipStream_t stream) {
  const float* img_f0 = (const float*)d_in[0];
  const float* img_f1 = (const float*)d_in[1];
  const float* img_f2 = (const float*)d_in[2];
  const float* img_f3 = (const float*)d_in[3];
  const float* img_f4 = (const float*)d_in[4];
  const float* pts_f0 = (const float*)d_in[5];
  const float* img_w    = (const float*)d_in[6];
  const float* img_b    = (const float*)d_in[7];
  const float* img_gam  = (const float*)d_in[8];
  const float* img_bet  = (const float*)d_in[9];
  const float* img_mean = (const float*)d_in[10];
  const float* img_var  = (const float*)d_in[11];
  const float* pts_w    = (const float*)d_in[12];
  const float* pts_b    = (const float*)d_in[13];
  const float* pts_gam  = (const float*)d_in[14];
  const float* pts_bet  = (const float*)d_in[15];
  const float* pts_mean = (const float*)d_in[16];
  const float* pts_var  = (const float*)d_in[17];

  float* ws = (float*)d_ws;
  float* img_m     = ws;                                   // 2*512*65536 floats (256 MiB)
  float* alpha_img = ws + (long)NBATCH * CCH * HWSZ;
  float* beta_img  = alpha_img + CCH;
  float* alpha_pts = beta_img + CCH;
  float* beta_pts  = alpha_pts + CCH;

  // 1) multi-level img feature sum
  long elems = (long)NBATCH * CCH * HWSZ;                  // 67,108,864
  build_img_m<<<(unsigned)(elems / 256), 256, 0, stream>>>(
      img_f0, img_f1, img_f2, img_f3, img_f4, img_m);

  // 2) BN folding
  bn_affine<<<2, 256, 0, stream>>>(img_b, img_gam, img_bet, img_mean, img_var,
                                   alpha_img, beta_img);
  bn_affine<<<2, 256, 0, stream>>>(pts_b, pts_gam, pts_bet, pts_mean, pts_var,
                                   alpha_pts, beta_pts);

  // 3) two WMMA GEMMs writing directly into the concatenated output
  dim3 grid(HWSZ / BNT, CCH / BM, NBATCH);                 // (512, 4, 2)
  float* out = (float*)d_out;
  gemm_bn_relu<<<grid, 256, 0, stream>>>(
      img_m, img_w, alpha_img, beta_img,
      out /* img half: channels [0,512) */,
      (long)CCH * HWSZ, (long)2 * CCH * HWSZ);
  gemm_bn_relu<<<grid, 256, 0, stream>>>(
      pts_f0, pts_w, alpha_pts, beta_pts,
      out + (long)CCH * HWSZ /* pts half: channels [512,1024) */,
      (long)CCH * HWSZ, (long)2 * CCH * HWSZ);
}